// PatchAttentionWithPairBias_20770461843492
// MI455X (gfx1250) — compile-verified
//
#include <hip/hip_runtime.h>
#include <hip/hip_bf16.h>

typedef float v2f __attribute__((ext_vector_type(2)));
typedef float v8f __attribute__((ext_vector_type(8)));

#define BB 4
#define NN 512
#define FDIM 256
#define ADIM 256
#define PDIM 128
#define DPD 64
#define LN_EPS 1e-3f

__device__ __forceinline__ float wsum16(float v) {
#pragma unroll
  for (int m = 1; m < 16; m <<= 1) v += __shfl_xor(v, m, 32);
  return v;
}
__device__ __forceinline__ float wsum32(float v) {
#pragma unroll
  for (int m = 1; m < 32; m <<= 1) v += __shfl_xor(v, m, 32);
  return v;
}
__device__ __forceinline__ float wmax32(float v) {
#pragma unroll
  for (int m = 1; m < 32; m <<= 1) v = fmaxf(v, __shfl_xor(v, m, 32));
  return v;
}

// ---------------- K0: fold pair-path weights: gw[64], sgw, c2 ----------------
__global__ void prep_kernel(const float* __restrict__ g_p, const float* __restrict__ b_p,
                            const float* __restrict__ W_pt, const float* __restrict__ b_pt,
                            const float* __restrict__ W_ph,
                            float* __restrict__ gw, float* __restrict__ consts) {
  __shared__ float s_gw[DPD], s_bw[DPD];
  int k = threadIdx.x;  // 64 threads
  if (k < DPD) {
    float w = 0.f;
    for (int p = 0; p < PDIM; ++p) w += W_pt[k * PDIM + p] * W_ph[p];
    s_gw[k] = g_p[k] * w;
    s_bw[k] = b_p[k] * w;
    gw[k] = s_gw[k];
  }
  __syncthreads();
  if (k == 0) {
    float sgw = 0.f, cb = 0.f;
    for (int i = 0; i < DPD; ++i) { sgw += s_gw[i]; cb += s_bw[i]; }
    for (int p = 0; p < PDIM; ++p) cb += b_pt[p] * W_ph[p];
    consts[0] = sgw;
    consts[1] = cb;
  }
}

// ---------------- K1: LayerNorm over F rows (wave per row of 256) ----------------
__global__ void lnF_kernel(const float* __restrict__ F, const float* __restrict__ g_f,
                           const float* __restrict__ b_f, float* __restrict__ Fn) {
  int lane = threadIdx.x & 31;
  int wid = threadIdx.x >> 5;
  int row = blockIdx.x * 8 + wid;  // 2048 rows
  const float* x = F + (size_t)row * FDIM;
  int c = lane * 8;
  float4 a = *(const float4*)(x + c);
  float4 b = *(const float4*)(x + c + 4);
  float s = a.x + a.y + a.z + a.w + b.x + b.y + b.z + b.w;
  float s2 = a.x * a.x + a.y * a.y + a.z * a.z + a.w * a.w +
             b.x * b.x + b.y * b.y + b.z * b.z + b.w * b.w;
  s = wsum32(s);
  s2 = wsum32(s2);
  float mu = s * (1.0f / FDIM);
  float var = s2 * (1.0f / FDIM) - mu * mu;
  float rs = rsqrtf(var + LN_EPS);
  float4 ga = *(const float4*)(g_f + c), gb = *(const float4*)(g_f + c + 4);
  float4 ba = *(const float4*)(b_f + c), bb = *(const float4*)(b_f + c + 4);
  float4 oa, ob;
  oa.x = (a.x - mu) * rs * ga.x + ba.x;
  oa.y = (a.y - mu) * rs * ga.y + ba.y;
  oa.z = (a.z - mu) * rs * ga.z + ba.z;
  oa.w = (a.w - mu) * rs * ga.w + ba.w;
  ob.x = (b.x - mu) * rs * gb.x + bb.x;
  ob.y = (b.y - mu) * rs * gb.y + bb.y;
  ob.z = (b.z - mu) * rs * gb.z + bb.z;
  ob.w = (b.w - mu) * rs * gb.w + bb.w;
  float* y = Fn + (size_t)row * FDIM;
  *(float4*)(y + c) = oa;
  *(float4*)(y + c + 4) = ob;
}

// ---------------- K2: Q/K/V = Fn @ W via v_wmma_f32_16x16x4_f32 ----------------
// one wave computes a 16x16 tile; grid (128 mtiles, 4 ngroups, 3 matrices), block 128
__global__ void qkv_wmma_kernel(const float* __restrict__ Fn,
                                const float* __restrict__ Wq, const float* __restrict__ Wk,
                                const float* __restrict__ Wv,
                                float* __restrict__ Q, float* __restrict__ K,
                                float* __restrict__ V) {
  const float* W;
  float* O;
  if (blockIdx.z == 0) { W = Wq; O = Q; }
  else if (blockIdx.z == 1) { W = Wk; O = K; }
  else { W = Wv; O = V; }
  int lane = threadIdx.x & 31;
  int wid = threadIdx.x >> 5;
  int m0 = blockIdx.x * 16;
  int n0 = (blockIdx.y * 4 + wid) * 16;
  int lm = lane & 15, lh = lane >> 4;
  const float* arow = Fn + (size_t)(m0 + lm) * FDIM + 2 * lh;
  v8f acc = {};
  for (int k0 = 0; k0 < FDIM; k0 += 4) {
    v2f a = *(const v2f*)(arow + k0);  // A[M=lm][K=k0+2lh, k0+2lh+1]
    int kr = k0 + 2 * lh;
    v2f b;
    b.x = W[(size_t)kr * ADIM + n0 + lm];        // B[K=kr][N=lm]
    b.y = W[(size_t)(kr + 1) * ADIM + n0 + lm];  // B[K=kr+1][N=lm]
    acc = __builtin_amdgcn_wmma_f32_16x16x4_f32(false, a, false, b, (short)0, acc,
                                                false, false);
  }
  int col = n0 + lm;
  int rbase = m0 + (lh ? 8 : 0);
#pragma unroll
  for (int i = 0; i < 8; ++i) O[(size_t)(rbase + i) * ADIM + col] = acc[i];
}

// ---------------- K3: vsum[b,k] = sum_d V[b,k,d] (wave per row) ----------------
__global__ void vsum_kernel(const float* __restrict__ V, float* __restrict__ vsum) {
  int lane = threadIdx.x & 31;
  int wid = threadIdx.x >> 5;
  int row = blockIdx.x * 8 + wid;  // 2048 rows
  const float* x = V + (size_t)row * ADIM;
  int c = lane * 8;
  float4 a = *(const float4*)(x + c);
  float4 b = *(const float4*)(x + c + 4);
  float s = a.x + a.y + a.z + a.w + b.x + b.y + b.z + b.w;
  s = wsum32(s);
  if (lane == 0) vsum[row] = s;
}

// ---------------- K4: bias[b,i,j] = rs*(sum d*gw - mu*sgw) + c2 ----------------
// Bandwidth kernel: wave handles 2 rows/iter, lane loads float4 (global_load_b128).
__global__ void bias_kernel(const float* __restrict__ D, const float* __restrict__ gw,
                            const float* __restrict__ consts, float* __restrict__ bias) {
  __shared__ float s_gw[DPD];
  if (threadIdx.x < DPD) s_gw[threadIdx.x] = gw[threadIdx.x];
  __syncthreads();
  float sgw = consts[0], c2 = consts[1];
  int lane = threadIdx.x & 31;
  int wid = threadIdx.x >> 5;
  int lm = lane & 15, lh = lane >> 4;
  float g0 = s_gw[lm * 4 + 0], g1 = s_gw[lm * 4 + 1];
  float g2 = s_gw[lm * 4 + 2], g3 = s_gw[lm * 4 + 3];
  const long ROWS = (long)BB * NN * NN;  // 1,048,576
  long wavesPerGrid = (long)gridDim.x * (blockDim.x >> 5);
  long wgid = (long)blockIdx.x * (blockDim.x >> 5) + wid;
  long stride = wavesPerGrid * 2;
  for (long r0 = wgid * 2; r0 < ROWS; r0 += stride) {
    long row = r0 + lh;
    const float* p = D + row * DPD + lm * 4;
    __builtin_prefetch(p + stride * DPD, 0, 1);  // global_prefetch_b8, next grid-stride
    float4 x = *(const float4*)p;
    float sd = x.x + x.y + x.z + x.w;
    float sd2 = x.x * x.x + x.y * x.y + x.z * x.z + x.w * x.w;
    float sg = x.x * g0 + x.y * g1 + x.z * g2 + x.w * g3;
    sd = wsum16(sd);
    sd2 = wsum16(sd2);
    sg = wsum16(sg);
    if (lm == 0) {
      float mu = sd * (1.0f / DPD);
      float var = sd2 * (1.0f / DPD) - mu * mu;
      float rs = rsqrtf(var + LN_EPS);
      bias[row] = rs * (sg - mu * sgw) + c2;
    }
  }
}

// ---------------- K5: logits = Q @ K^T per batch, WMMA ----------------
__global__ void qk_wmma_kernel(const float* __restrict__ Q, const float* __restrict__ Km,
                               float* __restrict__ logits) {
  int b = blockIdx.z;
  int lane = threadIdx.x & 31;
  int wid = threadIdx.x >> 5;
  int q0 = blockIdx.x * 16;
  int k0t = (blockIdx.y * 4 + wid) * 16;
  int lm = lane & 15, lh = lane >> 4;
  const float* Qb = Q + (size_t)b * NN * ADIM;
  const float* Kb = Km + (size_t)b * NN * ADIM;
  const float* arow = Qb + (size_t)(q0 + lm) * ADIM + 2 * lh;
  const float* brow = Kb + (size_t)(k0t + lm) * ADIM + 2 * lh;  // B = K^T
  v8f acc = {};
  for (int kd = 0; kd < ADIM; kd += 4) {
    v2f a = *(const v2f*)(arow + kd);
    v2f bv = *(const v2f*)(brow + kd);
    acc = __builtin_amdgcn_wmma_f32_16x16x4_f32(false, a, false, bv, (short)0, acc,
                                                false, false);
  }
  float* Lb = logits + (size_t)b * NN * NN;
  int col = k0t + lm;
  int rbase = q0 + (lh ? 8 : 0);
#pragma unroll
  for (int i = 0; i < 8; ++i) Lb[(size_t)(rbase + i) * NN + col] = acc[i];
}

// ---------------- K6: out[b,q] = softmax(s*(logit+bias)) . vsum[b,:] ----------------
__global__ void softmax_out_kernel(const float* __restrict__ logits,
                                   const float* __restrict__ bias,
                                   const float* __restrict__ vsum,
                                   float* __restrict__ out) {
  int lane = threadIdx.x & 31;
  int wid = threadIdx.x >> 5;
  int row = blockIdx.x * 8 + wid;  // b*N + q, 2048 rows
  int b = row >> 9;
  const float* l = logits + (size_t)row * NN;
  const float* bi = bias + (size_t)row * NN;
  const float* vs = vsum + (size_t)b * NN;
  const float scalar = 0.0625f;  // 1/sqrt(AD=256)
  float4 z[4], v[4];
  float m = -INFINITY;
#pragma unroll
  for (int i = 0; i < 4; ++i) {
    int c = i * 128 + lane * 4;
    float4 lv = *(const float4*)(l + c);
    float4 bv = *(const float4*)(bi + c);
    v[i] = *(const float4*)(vs + c);
    z[i].x = scalar * (lv.x + bv.x);
    z[i].y = scalar * (lv.y + bv.y);
    z[i].z = scalar * (lv.z + bv.z);
    z[i].w = scalar * (lv.w + bv.w);
    m = fmaxf(m, fmaxf(fmaxf(z[i].x, z[i].y), fmaxf(z[i].z, z[i].w)));
  }
  m = wmax32(m);
  float se = 0.f, sv = 0.f;
#pragma unroll
  for (int i = 0; i < 4; ++i) {
    float e0 = __expf(z[i].x - m);
    float e1 = __expf(z[i].y - m);
    float e2 = __expf(z[i].z - m);
    float e3 = __expf(z[i].w - m);
    se += e0 + e1 + e2 + e3;
    sv += e0 * v[i].x + e1 * v[i].y + e2 * v[i].z + e3 * v[i].w;
  }
  se = wsum32(se);
  sv = wsum32(sv);
  if (lane == 0) out[row] = sv / se;
}

extern "C" void kernel_launch(void* const* d_in, const int* in_sizes, int n_in,
                              void* d_out, int out_size, void* d_ws, size_t ws_size,
                              hipStream_t stream) {
  const float* F = (const float*)d_in[0];
  const float* D = (const float*)d_in[1];
  const float* Wq = (const float*)d_in[2];
  const float* Wk = (const float*)d_in[3];
  const float* Wv = (const float*)d_in[4];
  const float* g_f = (const float*)d_in[5];
  const float* b_f = (const float*)d_in[6];
  const float* g_p = (const float*)d_in[7];
  const float* b_p = (const float*)d_in[8];
  const float* W_pt = (const float*)d_in[9];
  const float* b_pt = (const float*)d_in[10];
  const float* W_ph = (const float*)d_in[11];
  float* out = (float*)d_out;

  // workspace layout (floats)
  float* ws = (float*)d_ws;
  float* gw = ws;                 // 64
  float* consts = ws + 64;        // 2 (pad to 128)
  float* Fn = ws + 128;           // 2048*256 = 524288
  float* Q = Fn + 524288;         // 524288
  float* K = Q + 524288;          // 524288
  float* V = K + 524288;          // 524288
  float* vsum = V + 524288;       // 2048
  float* bias = vsum + 2048;      // 1048576
  float* logits = bias + 1048576; // 1048576

  prep_kernel<<<1, 64, 0, stream>>>(g_p, b_p, W_pt, b_pt, W_ph, gw, consts);
  lnF_kernel<<<256, 256, 0, stream>>>(F, g_f, b_f, Fn);
  qkv_wmma_kernel<<<dim3(128, 4, 3), 128, 0, stream>>>(Fn, Wq, Wk, Wv, Q, K, V);
  vsum_kernel<<<256, 256, 0, stream>>>(V, vsum);
  bias_kernel<<<4096, 256, 0, stream>>>(D, gw, consts, bias);
  qk_wmma_kernel<<<dim3(32, 8, 4), 128, 0, stream>>>(Q, K, logits);
  softmax_out_kernel<<<256, 256, 0, stream>>>(logits, bias, vsum, out);
}